// RWKV_TimeMix_41755672052465
// MI455X (gfx1250) — compile-verified
//
#include <hip/hip_runtime.h>
#include <hip/hip_bf16.h>
#include <math.h>

typedef float v2f __attribute__((ext_vector_type(2)));
typedef float v8f __attribute__((ext_vector_type(8)));

#define B_ 4
#define T_ 1024
#define C_ 512
#define D_ 16
#define O_ 64

// ---------------------------------------------------------------------------
// Kernel 1: w_sum[u] = alpha[u] * sum_{j=0}^{T-1-u} time_w[T-1-j] * beta[u+j]
// One block per u, 256-thread strided partial sums + LDS tree reduction.
// ---------------------------------------------------------------------------
__global__ __launch_bounds__(256) void wsum_kernel(
    const float* __restrict__ time_w,   // [T]
    const float* __restrict__ alpha,    // [T]  (H=1, [1,1,TT])
    const float* __restrict__ beta,     // [T]  (H=1, [1,TT,1])
    float* __restrict__ wsum)           // [T]
{
  __shared__ float red[256];
  const int u = blockIdx.x;
  const int tid = threadIdx.x;
  const int n = T_ - u;
  float acc = 0.f;
  for (int j = tid; j < n; j += 256)
    acc += time_w[T_ - 1 - j] * beta[u + j];
  red[tid] = acc;
  __syncthreads();
  for (int s = 128; s > 0; s >>= 1) {
    if (tid < s) red[tid] += red[tid + s];
    __syncthreads();
  }
  if (tid == 0) wsum[u] = alpha[u] * red[0];
}

// ---------------------------------------------------------------------------
// Kernel 2: k/v/r projections via V_WMMA_F32_16X16X4_F32.
// One wave per block, one 16-row (b*T+t) tile per wave, all 16 D outputs.
// Time shift (first 256 channels, shift right by 1 in t, zero at t=0) is
// folded into the A-matrix load.
// ---------------------------------------------------------------------------
__global__ __launch_bounds__(32) void proj_kernel(
    const float* __restrict__ x,        // [B,T,C]
    const float* __restrict__ Wk, const float* __restrict__ bk,
    const float* __restrict__ Wv, const float* __restrict__ bv,
    const float* __restrict__ Wr, const float* __restrict__ br,
    float* __restrict__ k_out,          // [B*T, D]
    float* __restrict__ kv_out,         // [B*T, D]
    float* __restrict__ r_out)          // [B*T, D]
{
  const int lane = threadIdx.x;         // 0..31
  const int row0 = blockIdx.x * 16;     // tile base row in flattened [B*T]
  const int m    = lane & 15;           // A-matrix row within tile
  const int half = lane >> 4;           // K sub-pair select
  const int nn   = lane & 15;           // B/D column (D index)
  const int row  = row0 + m;
  const int t    = row & (T_ - 1);
  const int koff = 2 * half;

  const float* pnorm  = x + (size_t)row * C_;
  const float* pshift = (t == 0) ? pnorm : (pnorm - C_); // keep in-bounds

  v8f ak = {}, av = {}, ar = {};

  // First half of channels: time-shifted (zero at t==0)
  for (int kb = 0; kb < C_ / 2; kb += 4) {
    const int c = kb + koff;
    v2f a = *(const v2f*)(pshift + c);
    if (t == 0) { a.x = 0.f; a.y = 0.f; }
    v2f wk2 = *(const v2f*)(Wk + (size_t)nn * C_ + c);
    v2f wv2 = *(const v2f*)(Wv + (size_t)nn * C_ + c);
    v2f wr2 = *(const v2f*)(Wr + (size_t)nn * C_ + c);
    ak = __builtin_amdgcn_wmma_f32_16x16x4_f32(false, a, false, wk2, (short)0, ak, false, false);
    av = __builtin_amdgcn_wmma_f32_16x16x4_f32(false, a, false, wv2, (short)0, av, false, false);
    ar = __builtin_amdgcn_wmma_f32_16x16x4_f32(false, a, false, wr2, (short)0, ar, false, false);
  }
  // Second half of channels: unshifted
  for (int kb = C_ / 2; kb < C_; kb += 4) {
    const int c = kb + koff;
    v2f a = *(const v2f*)(pnorm + c);
    v2f wk2 = *(const v2f*)(Wk + (size_t)nn * C_ + c);
    v2f wv2 = *(const v2f*)(Wv + (size_t)nn * C_ + c);
    v2f wr2 = *(const v2f*)(Wr + (size_t)nn * C_ + c);
    ak = __builtin_amdgcn_wmma_f32_16x16x4_f32(false, a, false, wk2, (short)0, ak, false, false);
    av = __builtin_amdgcn_wmma_f32_16x16x4_f32(false, a, false, wv2, (short)0, av, false, false);
    ar = __builtin_amdgcn_wmma_f32_16x16x4_f32(false, a, false, wr2, (short)0, ar, false, false);
  }

  const float bkv = bk[nn], bvv = bv[nn], brv = br[nn];
#pragma unroll
  for (int j = 0; j < 8; ++j) {
    const int rr = row0 + j + 8 * half;   // D-matrix: VGPR j -> M = j + 8*half
    float kk = ak[j] + bkv;
    kk = __expf(fminf(fmaxf(kk, -60.f), 30.f));
    const float vv = av[j] + bvv;
    const float rv = ar[j] + brv;
    k_out [(size_t)rr * D_ + nn] = kk;
    kv_out[(size_t)rr * D_ + nn] = kk * vv;
    r_out [(size_t)rr * D_ + nn] = rv;
  }
}

// ---------------------------------------------------------------------------
// Kernel 3: per (b,d) inclusive cumsum of k over t (Hillis-Steele in LDS),
// written back in place, plus mean over t of kv.
// ---------------------------------------------------------------------------
__global__ __launch_bounds__(1024) void scan_kernel(
    float* __restrict__ k_inout,        // [B*T, D] -> becomes cumsum
    const float* __restrict__ kv,       // [B*T, D]
    float* __restrict__ meankv)         // [B, D]
{
  __shared__ float buf[T_];
  __shared__ float red[T_];
  const int bd = blockIdx.x;            // 0..B*D-1
  const int b = bd >> 4, d = bd & 15;
  const int t = threadIdx.x;
  const size_t idx = ((size_t)(b * T_ + t)) * D_ + d;

  buf[t] = k_inout[idx];
  red[t] = kv[idx];
  __syncthreads();

  for (int off = 1; off < T_; off <<= 1) {
    const float add = (t >= off) ? buf[t - off] : 0.f;
    const float cur = buf[t];
    __syncthreads();
    buf[t] = cur + add;
    __syncthreads();
  }
  k_inout[idx] = buf[t];

  for (int s = T_ / 2; s > 0; s >>= 1) {
    if (t < s) red[t] += red[t + s];
    __syncthreads();
  }
  if (t == 0) meankv[b * D_ + d] = red[0] * (1.0f / T_);
}

// ---------------------------------------------------------------------------
// Kernel 4: rwkv = r * w_sum[t] * mean_kv[b,d] / (sum_k + 1e-8), then
// out = (rwkv @ Wo.T + bo) * gamma[t] via WMMA (M=16 rows, K=16=D, N=64).
// rwkv is computed inline inside the A-matrix load.
// ---------------------------------------------------------------------------
__global__ __launch_bounds__(32) void out_kernel(
    const float* __restrict__ r_in,     // [B*T, D]
    const float* __restrict__ sumk,     // [B*T, D]
    const float* __restrict__ wsum,     // [T]
    const float* __restrict__ meankv,   // [B, D]
    const float* __restrict__ Wo,       // [64, D]
    const float* __restrict__ bo,       // [64]
    const float* __restrict__ gamma,    // [T]
    float* __restrict__ out)            // [B*T, 64]
{
  const int lane = threadIdx.x;
  const int row0 = blockIdx.x * 16;
  const int m    = lane & 15;
  const int half = lane >> 4;
  const int nn   = lane & 15;
  const int row  = row0 + m;
  const int b    = row >> 10;
  const int t    = row & (T_ - 1);
  const float wst = wsum[t];
  const int koff = 2 * half;

  v8f a0 = {}, a1 = {}, a2 = {}, a3 = {};

#pragma unroll
  for (int kb = 0; kb < D_; kb += 4) {
    const int c = kb + koff;
    const v2f rr = *(const v2f*)(r_in   + (size_t)row * D_ + c);
    const v2f sk = *(const v2f*)(sumk   + (size_t)row * D_ + c);
    const v2f mk = *(const v2f*)(meankv + (size_t)b * D_ + c);
    v2f a;
    a.x = rr.x * wst * mk.x / (sk.x + 1e-8f);
    a.y = rr.y * wst * mk.y / (sk.y + 1e-8f);
    const v2f w0 = *(const v2f*)(Wo + (size_t)( 0 + nn) * D_ + c);
    const v2f w1 = *(const v2f*)(Wo + (size_t)(16 + nn) * D_ + c);
    const v2f w2 = *(const v2f*)(Wo + (size_t)(32 + nn) * D_ + c);
    const v2f w3 = *(const v2f*)(Wo + (size_t)(48 + nn) * D_ + c);
    a0 = __builtin_amdgcn_wmma_f32_16x16x4_f32(false, a, false, w0, (short)0, a0, false, false);
    a1 = __builtin_amdgcn_wmma_f32_16x16x4_f32(false, a, false, w1, (short)0, a1, false, false);
    a2 = __builtin_amdgcn_wmma_f32_16x16x4_f32(false, a, false, w2, (short)0, a2, false, false);
    a3 = __builtin_amdgcn_wmma_f32_16x16x4_f32(false, a, false, w3, (short)0, a3, false, false);
  }

  const float b0 = bo[ 0 + nn], b1 = bo[16 + nn], b2 = bo[32 + nn], b3 = bo[48 + nn];
#pragma unroll
  for (int j = 0; j < 8; ++j) {
    const int rr = row0 + j + 8 * half;
    const float g = gamma[rr & (T_ - 1)];
    float* orow = out + (size_t)rr * O_;
    orow[ 0 + nn] = (a0[j] + b0) * g;
    orow[16 + nn] = (a1[j] + b1) * g;
    orow[32 + nn] = (a2[j] + b2) * g;
    orow[48 + nn] = (a3[j] + b3) * g;
  }
}

// ---------------------------------------------------------------------------
extern "C" void kernel_launch(void* const* d_in, const int* in_sizes, int n_in,
                              void* d_out, int out_size, void* d_ws, size_t ws_size,
                              hipStream_t stream) {
  (void)in_sizes; (void)n_in; (void)out_size; (void)ws_size;
  const float* x  = (const float*)d_in[0];
  const float* tw = (const float*)d_in[1];   // time_w    [1,1024]
  const float* ta = (const float*)d_in[2];   // time_alpha[1,1,1024]
  const float* tb = (const float*)d_in[3];   // time_beta [1,1024,1]
  const float* tg = (const float*)d_in[4];   // time_gamma[1024,1]
  const float* Wk = (const float*)d_in[5];
  const float* bk = (const float*)d_in[6];
  const float* Wv = (const float*)d_in[7];
  const float* bv = (const float*)d_in[8];
  const float* Wr = (const float*)d_in[9];
  const float* br = (const float*)d_in[10];
  const float* Wo = (const float*)d_in[11];
  const float* bo = (const float*)d_in[12];
  float* out = (float*)d_out;

  float* ws     = (float*)d_ws;
  float* wsum   = ws;                        // 1024
  float* kbuf   = wsum + T_;                 // 65536 (becomes cumsum in place)
  float* kvbuf  = kbuf + (size_t)B_ * T_ * D_;
  float* rbuf   = kvbuf + (size_t)B_ * T_ * D_;
  float* mkv    = rbuf + (size_t)B_ * T_ * D_;  // B*D = 64

  wsum_kernel<<<T_, 256, 0, stream>>>(tw, ta, tb, wsum);
  proj_kernel<<<(B_ * T_) / 16, 32, 0, stream>>>(x, Wk, bk, Wv, bv, Wr, br,
                                                 kbuf, kvbuf, rbuf);
  scan_kernel<<<B_ * D_, 1024, 0, stream>>>(kbuf, kvbuf, mkv);
  out_kernel<<<(B_ * T_) / 16, 32, 0, stream>>>(rbuf, kbuf, wsum, mkv,
                                                Wo, bo, tg, out);
}